// PromptDetectionLoss_20109036880352
// MI455X (gfx1250) — compile-verified
//
#include <hip/hip_runtime.h>
#include <math.h>

#define TOPK 13
#define REG_MAX 16
#define EPSF 1e-7f

typedef unsigned int u32;
typedef unsigned long long u64;

typedef __attribute__((ext_vector_type(2))) float v2f;
typedef __attribute__((ext_vector_type(8))) float v8f;
typedef int v2i32 __attribute__((vector_size(8)));
typedef int v4i32 __attribute__((vector_size(16)));

// ---------------- CDNA5 feature probes (device pass only) ----------------
#if defined(__HIP_DEVICE_COMPILE__) && __has_builtin(__builtin_amdgcn_global_load_async_to_lds_b128) && __has_builtin(__builtin_amdgcn_global_load_async_to_lds_b64) && __has_builtin(__builtin_amdgcn_global_load_async_to_lds_b32)
#define USE_ASYNC 1
#else
#define USE_ASYNC 0
#endif

#if defined(__HIP_DEVICE_COMPILE__) && __has_builtin(__builtin_amdgcn_wmma_f32_16x16x4_f32)
#define USE_WMMA 1
#else
#define USE_WMMA 0
#endif

#if USE_ASYNC
#define ASG __attribute__((address_space(1)))
#define ASL __attribute__((address_space(3)))
__device__ __forceinline__ void async_b128(const void* g, void* l) {
  __builtin_amdgcn_global_load_async_to_lds_b128((ASG v4i32*)g, (ASL v4i32*)l, 0, 0);
}
__device__ __forceinline__ void async_b64(const void* g, void* l) {
  __builtin_amdgcn_global_load_async_to_lds_b64((ASG v2i32*)g, (ASL v2i32*)l, 0, 0);
}
__device__ __forceinline__ void async_b32(const void* g, void* l) {
  __builtin_amdgcn_global_load_async_to_lds_b32((ASG int*)g, (ASL int*)l, 0, 0);
}
__device__ __forceinline__ void async_wait0() {
#if __has_builtin(__builtin_amdgcn_s_wait_asynccnt)
  __builtin_amdgcn_s_wait_asynccnt(0);
#else
  asm volatile("s_wait_asynccnt 0x0" ::: "memory");
#endif
}
#endif

// ---------------- helpers ----------------
__device__ __forceinline__ u32 ordf(float f) {
  u32 u = __float_as_uint(f);
  return (u & 0x80000000u) ? ~u : (u | 0x80000000u);
}
__device__ __forceinline__ float unordf(u32 u) {
  return __uint_as_float((u & 0x80000000u) ? (u & 0x7fffffffu) : ~u);
}
__device__ __forceinline__ float sigmoidf_(float x) { return 1.0f / (1.0f + expf(-x)); }

__device__ __forceinline__ float iou4(float a0, float a1, float a2, float a3,
                                      float b0, float b1, float b2, float b3) {
  float x1 = fmaxf(a0, b0), y1 = fmaxf(a1, b1);
  float x2 = fminf(a2, b2), y2 = fminf(a3, b3);
  float inter = fmaxf(x2 - x1, 0.f) * fmaxf(y2 - y1, 0.f);
  float aa = (a2 - a0) * (a3 - a1);
  float ab = (b2 - b0) * (b3 - b1);
  return inter / (aa + ab - inter + EPSF);
}

// bestkey init: value=-2.0 (ordered 0x3FFFFFFF), g encoded as ~g with g=0
#define BESTKEY_INIT ((((u64)0x3FFFFFFFu) << 32) | 0xFFFFFFFFull)

// ---------------- kernel 0: workspace init ----------------
__global__ __launch_bounds__(256) void init_kernel(u64* __restrict__ bestkey,
                                                   float* __restrict__ partial,
                                                   long long nkeys, long long npart) {
  long long i = (long long)blockIdx.x * 256 + threadIdx.x;
  if (i < nkeys) bestkey[i] = BESTKEY_INIT;
  if (i < npart) partial[i] = 0.0f;
}

// ---------------- kernel 1: per-(image,gt) top-13 + scatter ----------------
__global__ __launch_bounds__(256) void topk_kernel(const float* __restrict__ boxes,
                                                   const float* __restrict__ scores,
                                                   const float* __restrict__ anchors,
                                                   const float* __restrict__ gts,
                                                   u64* __restrict__ bestkey,
                                                   int N, int G) {
  const int tid = threadIdx.x;
  const int b = blockIdx.x / G;
  const int g = blockIdx.x % G;
  const float* gt = gts + (size_t)(b * G + g) * 4;
  const float gx0 = gt[0], gy0 = gt[1], gx1 = gt[2], gy1 = gt[3];
  const float* bx = boxes + (size_t)b * N * 4;
  const float* sc = scores + (size_t)b * N;

  __shared__ float s_box[256 * 4];
  __shared__ float s_anc[256 * 2];
  __shared__ float s_sco[256];
  __shared__ u64 s_cand[256 * TOPK];
  __shared__ u64 s_red[256];

  float lv[TOPK];
  int li[TOPK];
#pragma unroll
  for (int j = 0; j < TOPK; ++j) { lv[j] = -2.0f; li[j] = 0; }

  const int tiles = (N + 255) / 256;
  for (int t = 0; t < tiles; ++t) {
    int idx = t * 256 + tid;
    __syncthreads();  // previous tile fully consumed
#if USE_ASYNC
    if (idx < N) {
      async_b128(&bx[(size_t)idx * 4], &s_box[tid * 4]);
      async_b64(&anchors[(size_t)idx * 2], &s_anc[tid * 2]);
      async_b32(&sc[idx], &s_sco[tid]);
    }
    async_wait0();
#else
    if (idx < N) {
      s_box[tid * 4 + 0] = bx[(size_t)idx * 4 + 0];
      s_box[tid * 4 + 1] = bx[(size_t)idx * 4 + 1];
      s_box[tid * 4 + 2] = bx[(size_t)idx * 4 + 2];
      s_box[tid * 4 + 3] = bx[(size_t)idx * 4 + 3];
      s_anc[tid * 2 + 0] = anchors[(size_t)idx * 2 + 0];
      s_anc[tid * 2 + 1] = anchors[(size_t)idx * 2 + 1];
      s_sco[tid] = sc[idx];
    }
#endif
    __syncthreads();
    if (idx < N) {
      float a0 = s_box[tid * 4 + 0], a1 = s_box[tid * 4 + 1];
      float a2 = s_box[tid * 4 + 2], a3 = s_box[tid * 4 + 3];
      float ax = s_anc[tid * 2 + 0], ay = s_anc[tid * 2 + 1];
      float s = s_sco[tid];
      bool inside = (ax >= gx0) & (ax <= gx1) & (ay >= gy0) & (ay <= gy1);
      float align = -1.0f;
      if (inside) {
        float iou = iou4(a0, a1, a2, a3, gx0, gy0, gx1, gy1);
        float p = sigmoidf_(s);
        float i2 = iou * iou;
        align = p * i2 * i2 * i2;  // ALPHA=1, BETA=6
      }
      if (align > lv[TOPK - 1]) {
        lv[TOPK - 1] = align;
        li[TOPK - 1] = idx;
#pragma unroll
        for (int j = TOPK - 1; j > 0; --j) {
          if (lv[j] > lv[j - 1]) {
            float tv = lv[j]; lv[j] = lv[j - 1]; lv[j - 1] = tv;
            int ti = li[j]; li[j] = li[j - 1]; li[j - 1] = ti;
          }
        }
      }
    }
  }

  // dump per-thread candidates (packed: ordered value | anchor idx)
#pragma unroll
  for (int j = 0; j < TOPK; ++j)
    s_cand[tid * TOPK + j] = (((u64)ordf(lv[j])) << 32) | (u32)li[j];
  __syncthreads();

  // 13 rounds of block-wide max selection with destructive removal
  for (int r = 0; r < TOPK; ++r) {
    u64 lm = 0;
#pragma unroll
    for (int j = 0; j < TOPK; ++j) {
      u64 c = s_cand[tid * TOPK + j];
      lm = (c > lm) ? c : lm;
    }
    s_red[tid] = lm;
    __syncthreads();
    for (int off = 128; off > 0; off >>= 1) {
      if (tid < off) {
        u64 o = s_red[tid + off];
        if (o > s_red[tid]) s_red[tid] = o;
      }
      __syncthreads();
    }
    u64 w = s_red[0];
    if (w != 0ull) {
#pragma unroll
      for (int j = 0; j < TOPK; ++j)
        if (s_cand[tid * TOPK + j] == w) s_cand[tid * TOPK + j] = 0ull;
      if (tid == 0) {
        u32 anchor = (u32)w;
        u64 key = (w & 0xFFFFFFFF00000000ull) | (u64)(u32)(~(u32)g);
        atomicMax(&bestkey[(size_t)b * N + anchor], key);
      }
    }
    __syncthreads();
  }
}

// ---------------- kernel 2: per-anchor losses -> per-block partials ----------------
__global__ __launch_bounds__(256) void loss_kernel(const float* __restrict__ boxes,
                                                   const float* __restrict__ scores,
                                                   const float* __restrict__ anchors,
                                                   const float* __restrict__ strides,
                                                   const float* __restrict__ logits,
                                                   const float* __restrict__ prompt,
                                                   const float* __restrict__ gts,
                                                   const int* __restrict__ img_size_p,
                                                   const u64* __restrict__ bestkey,
                                                   float* __restrict__ partial,
                                                   int N, int G, int PD, int nblkp) {
  const int tid = threadIdx.x;
  const int b = blockIdx.y;
  const int n = blockIdx.x * 256 + tid;

  float acc[8];
#pragma unroll
  for (int q = 0; q < 8; ++q) acc[q] = 0.0f;

  if (n < N) {
    u64 key = bestkey[(size_t)b * N + n];
    float best = unordf((u32)(key >> 32));
    int g = (int)(~(u32)key);
    bool fg = (best >= 0.0f);

    float s = scores[(size_t)b * N + n];
    float prob = sigmoidf_(s);
    const float* bp = boxes + ((size_t)b * N + n) * 4;
    float a0 = bp[0], a1 = bp[1], a2 = bp[2], a3 = bp[3];

    float t_score = 0.0f, overlap = 0.0f;
    float tb0 = 0.f, tb1 = 0.f, tb2 = 0.f, tb3 = 0.f;
    if (fg) {
      const float* gt = gts + (size_t)(b * G + g) * 4;
      tb0 = gt[0]; tb1 = gt[1]; tb2 = gt[2]; tb3 = gt[3];
      overlap = iou4(a0, a1, a2, a3, tb0, tb1, tb2, tb3);
      t_score = fmaxf(overlap, 0.1f);
    }
    // BCE-with-logits over all anchors
    acc[0] = fmaxf(s, 0.f) - s * t_score + log1pf(expf(-fabsf(s)));

    if (fg) {
      // CIoU
      float iou = overlap;
      float cw = fmaxf(a2, tb2) - fminf(a0, tb0);
      float ch = fmaxf(a3, tb3) - fminf(a1, tb1);
      float c2 = cw * cw + ch * ch + EPSF;
      float dx = tb0 + tb2 - a0 - a2;
      float dy = tb1 + tb3 - a1 - a3;
      float rho2 = (dx * dx + dy * dy) * 0.25f;
      float w1 = a2 - a0, h1 = a3 - a1 + EPSF;
      float w2 = tb2 - tb0, h2 = tb3 - tb1 + EPSF;
      float dv = atanf(w2 / h2) - atanf(w1 / h1);
      float v = 0.405284734569351f * dv * dv;  // 4/pi^2
      float alpha = v / (v - iou + (1.0f + EPSF));
      float ciou = iou - rho2 / c2 - v * alpha;
      acc[1] = 1.0f - ciou;

      // DFL (only fg anchors ever read the 64 logits)
      float ax = anchors[(size_t)n * 2 + 0], ay = anchors[(size_t)n * 2 + 1];
      float st = strides[n];
      float d[4];
      d[0] = (ax - tb0) / st; d[1] = (ay - tb1) / st;
      d[2] = (tb2 - ax) / st; d[3] = (tb3 - ay) / st;
      const float* lg = logits + ((size_t)b * N + n) * (4 * REG_MAX);
      float dfl = 0.0f;
#pragma unroll
      for (int sd = 0; sd < 4; ++sd) {
        float di = fminf(fmaxf(d[sd], 0.0f), (float)REG_MAX - 1.0f - 0.01f);
        int tl = (int)di;
        int tr = tl + 1; if (tr > REG_MAX - 1) tr = REG_MAX - 1;
        float wl = (float)tr - di;
        float wr = 1.0f - wl;
        const float* L = lg + sd * REG_MAX;
        float m = -3.4e38f;
#pragma unroll
        for (int k = 0; k < REG_MAX; ++k) m = fmaxf(m, L[k]);
        float se = 0.0f;
#pragma unroll
        for (int k = 0; k < REG_MAX; ++k) se += expf(L[k] - m);
        float lse = m + logf(se);
        dfl += (lse - L[tl]) * wl + (lse - L[tr]) * wr;
      }
      acc[2] = dfl;

      // contrast
      float img = fmaxf((float)(*img_size_p), 1.0f);
      float cx = (tb0 + tb2) * 0.5f / img;
      float cy = (tb1 + tb3) * 0.5f / img;
      float cn = fmaxf(sqrtf(cx * cx + cy * cy), 1e-12f);
      cx /= cn; cy /= cn;
      float p0 = prompt[(size_t)b * PD + 0];
      float p1 = prompt[(size_t)b * PD + 1];
      float pn = fmaxf(sqrtf(p0 * p0 + p1 * p1), 1e-12f);
      p0 /= pn; p1 /= pn;
      acc[3] = 1.0f - (cx * p0 + cy * p1);

      acc[4] = 1.0f;     // n_pos
      acc[5] = prob;     // pos_score_sum
      acc[7] = overlap;  // matched_iou_sum
    } else {
      acc[6] = prob;     // neg_score_sum
    }
  }

  __shared__ float fred[256];
  for (int q = 0; q < 8; ++q) {
    fred[tid] = acc[q];
    __syncthreads();
    for (int off = 128; off > 0; off >>= 1) {
      if (tid < off) fred[tid] += fred[tid + off];
      __syncthreads();
    }
    if (tid == 0) partial[((size_t)b * nblkp + blockIdx.x) * 8 + q] = fred[0];
    __syncthreads();
  }
}

// ---------------- kernel 3: WMMA segmented reduction + final combine ----------------
// Sums partial[img][blk][q] over blk for each of 16 images via
// C += A(16x4) x ones(4x16)  (V_WMMA_F32_16X16X4_F32), then combines scalars.
__global__ __launch_bounds__(32) void final_kernel(const float* __restrict__ partial,
                                                   float* __restrict__ out,
                                                   int N, int B, int nblkp) {
  __shared__ float simg[8 * 16];
  const int lane = threadIdx.x;
#if USE_WMMA
  const int half = lane >> 4;   // 0: K=0,1   1: K=2,3
  const int m = lane & 15;      // image index (M row of A)
  v2f bones; bones.x = 1.0f; bones.y = 1.0f;
  for (int q = 0; q < 8; ++q) {
    v8f acc = {0.f, 0.f, 0.f, 0.f, 0.f, 0.f, 0.f, 0.f};
    for (int c = 0; c < nblkp; c += 4) {
      int k0 = c + half * 2;
      v2f A;
      A.x = (m < B) ? partial[((size_t)m * nblkp + k0 + 0) * 8 + q] : 0.0f;
      A.y = (m < B) ? partial[((size_t)m * nblkp + k0 + 1) * 8 + q] : 0.0f;
      acc = __builtin_amdgcn_wmma_f32_16x16x4_f32(false, A, false, bones,
                                                  (short)0, acc, false, false);
    }
    // D layout: VGPR r, lanes 0-15 -> M=r ; lanes 16-31 -> M=r+8 (cols identical)
    if (lane == 0) {
#pragma unroll
      for (int r = 0; r < 8; ++r) simg[q * 16 + r] = acc[r];
    }
    if (lane == 16) {
#pragma unroll
      for (int r = 0; r < 8; ++r) simg[q * 16 + 8 + r] = acc[r];
    }
  }
#else
  if (lane < 16) {
    for (int q = 0; q < 8; ++q) {
      float ssum = 0.0f;
      if (lane < B)
        for (int c = 0; c < nblkp; ++c)
          ssum += partial[((size_t)lane * nblkp + c) * 8 + q];
      simg[q * 16 + lane] = ssum;
    }
  }
#endif
  __syncthreads();
  if (lane == 0) {
    float match_s = 0, iou_s = 0, dfl_s = 0, con_s = 0;
    float npos_t = 0, pos_t = 0, neg_t = 0, miou_t = 0;
    for (int b = 0; b < B; ++b) {
      float bce = simg[0 * 16 + b], cio = simg[1 * 16 + b];
      float dfl = simg[2 * 16 + b], con = simg[3 * 16 + b];
      float np = simg[4 * 16 + b], ps = simg[5 * 16 + b];
      float ns = simg[6 * 16 + b], mi = simg[7 * 16 + b];
      float den = fmaxf(np, 1.0f);
      match_s += bce / (float)N;
      iou_s += cio / den;
      dfl_s += dfl / (4.0f * den);
      con_s += con / den;
      npos_t += np; pos_t += ps; neg_t += ns; miou_t += mi;
    }
    float nb = (float)B;
    float negcnt = (float)B * (float)N - npos_t;
    out[0] = (1.0f * match_s + 7.5f * iou_s + 1.5f * dfl_s + 1.0f * con_s) / nb;
    out[1] = match_s / nb;
    out[2] = iou_s / nb;
    out[3] = dfl_s / nb;
    out[4] = con_s / nb;
    out[5] = npos_t;
    out[6] = negcnt;
    out[7] = pos_t / fmaxf(npos_t, 1.0f);
    out[8] = neg_t / fmaxf(negcnt, 1.0f);
    out[9] = miou_t / fmaxf(npos_t, 1.0f);
  }
}

// ---------------- host ----------------
extern "C" void kernel_launch(void* const* d_in, const int* in_sizes, int n_in,
                              void* d_out, int out_size, void* d_ws, size_t ws_size,
                              hipStream_t stream) {
  const float* boxes = (const float*)d_in[0];
  const float* scores = (const float*)d_in[1];
  const float* anchors = (const float*)d_in[2];
  const float* strides = (const float*)d_in[3];
  const float* logits = (const float*)d_in[4];
  const float* prompt = (const float*)d_in[5];
  const float* gts = (const float*)d_in[6];
  const int* imgsz = (const int*)d_in[7];

  const int N = in_sizes[2] / 2;
  const int B = in_sizes[1] / N;
  const int G = in_sizes[6] / (4 * B);
  const int PD = in_sizes[5] / B;

  u64* bestkey = (u64*)d_ws;
  size_t keybytes = (size_t)B * N * sizeof(u64);
  float* partial = (float*)((char*)d_ws + ((keybytes + 255) / 256) * 256);

  const int nblk = (N + 255) / 256;
  const int nblkp = ((nblk + 3) / 4) * 4;  // pad to WMMA K-chunks of 4
  const long long nkeys = (long long)B * N;
  const long long npart = (long long)B * nblkp * 8;

  long long initn = nkeys > npart ? nkeys : npart;
  init_kernel<<<(unsigned)((initn + 255) / 256), 256, 0, stream>>>(bestkey, partial, nkeys, npart);
  topk_kernel<<<B * G, 256, 0, stream>>>(boxes, scores, anchors, gts, bestkey, N, G);
  loss_kernel<<<dim3(nblk, B), 256, 0, stream>>>(boxes, scores, anchors, strides, logits,
                                                 prompt, gts, imgsz, bestkey, partial,
                                                 N, G, PD, nblkp);
  final_kernel<<<1, 32, 0, stream>>>(partial, (float*)d_out, N, B, nblkp);
}